// FlameHead_39135742001815
// MI455X (gfx1250) — compile-verified
//
#include <hip/hip_runtime.h>
#include <math.h>

typedef __attribute__((ext_vector_type(2))) float v2f;
typedef __attribute__((ext_vector_type(8))) float v8f;

#define B_ 1024
#define V_ 5023
#define F_ 9976
#define J_ 5
#define L_ 68
#define NS_ 100
#define NE_ 50
#define NCOLS (V_ * 3)      /* 15069 */
#define KP 152              /* 150 padded to multiple of 4 */
#define KPOSE 36
#define NTILE 48            /* 16 vertices * 3 channels */
#define BT 16               /* batch rows per wave */
#define NWAVES 4
#define SD_STRIDE 153       /* LDS row stride (odd -> good bank spread) */

#if __has_builtin(__builtin_amdgcn_global_load_async_to_lds_b32) && \
    __has_builtin(__builtin_amdgcn_s_wait_asynccnt)
#define USE_ASYNC_LDS 1
typedef __attribute__((address_space(1))) int g_as1_int;
typedef __attribute__((address_space(3))) int l_as3_int;
#else
#define USE_ASYNC_LDS 0
#endif

static __device__ inline v8f wmma_f32x4(v2f a, v2f b, v8f c) {
  // D = A(16x4) * B(4x16) + C on the fp32 matrix pipe
  return __builtin_amdgcn_wmma_f32_16x16x4_f32(
      /*neg_a=*/false, a, /*neg_b=*/false, b,
      /*c_mod=*/(short)0, c, /*reuse_a=*/false, /*reuse_b=*/false);
}

// ---------------------------------------------------------------------------
// K1: assemble padded betas (B x 152): [shape(100) | expr(50) | 0 0]
// ---------------------------------------------------------------------------
__global__ void betas_pad_kernel(const float* __restrict__ sp,
                                 const float* __restrict__ ep,
                                 float* __restrict__ betasP) {
  int idx = blockIdx.x * 256 + threadIdx.x;
  if (idx >= B_ * KP) return;
  int b = idx / KP, l = idx - b * KP;
  float v = 0.0f;
  if (l < NS_)            v = sp[b * NS_ + l];
  else if (l < NS_ + NE_) v = ep[b * NE_ + (l - NS_)];
  betasP[idx] = v;
}

// ---------------------------------------------------------------------------
// K0: JSD[j,c,l] = sum_v JR[j,v]*SD[v,c,l]  (5 x 3 x 152, padded)
//     baseJ[j,c] = sum_v JR[j,v]*vt[v,c]
// ---------------------------------------------------------------------------
__global__ void jsd_kernel(const float* __restrict__ JR,
                           const float* __restrict__ sd,
                           const float* __restrict__ vt,
                           float* __restrict__ JSD,
                           float* __restrict__ baseJ) {
  int j = blockIdx.x;
  int t = threadIdx.x;
  if (t < 450) {
    int c = t / 150, l = t - c * 150;
    float acc = 0.0f;
    for (int v = 0; v < V_; ++v)
      acc += JR[j * V_ + v] * sd[(v * 3 + c) * 150 + l];
    JSD[(j * 3 + c) * KP + l] = acc;
  } else if (t < 456) {
    int q = t - 450;                        // zero the K-pad entries
    JSD[(j * 3 + q / 2) * KP + 150 + (q & 1)] = 0.0f;
  } else if (t < 459) {
    int c = t - 456;
    float acc = 0.0f;
    for (int v = 0; v < V_; ++v)
      acc += JR[j * V_ + v] * vt[v * 3 + c];
    baseJ[j * 3 + c] = acc;
  }
}

// ---------------------------------------------------------------------------
// K2: per-batch joints, rodrigues, kinematic chain, pose_feat, A matrices
// ---------------------------------------------------------------------------
static __device__ inline void rodrigues1(float r0, float r1, float r2,
                                         float* R) {
  float p0 = r0 + 1e-8f, p1 = r1 + 1e-8f, p2 = r2 + 1e-8f;
  float ang = sqrtf(p0 * p0 + p1 * p1 + p2 * p2);
  float inv = 1.0f / ang;
  float ax = r0 * inv, ay = r1 * inv, az = r2 * inv;
  float s = sinf(ang), cc = cosf(ang);
  float K[9] = {0.0f, -az, ay, az, 0.0f, -ax, -ay, ax, 0.0f};
  float K2[9];
#pragma unroll
  for (int m = 0; m < 3; ++m)
#pragma unroll
    for (int n = 0; n < 3; ++n)
      K2[m * 3 + n] = K[m * 3 + 0] * K[0 * 3 + n] +
                      K[m * 3 + 1] * K[1 * 3 + n] +
                      K[m * 3 + 2] * K[2 * 3 + n];
#pragma unroll
  for (int e = 0; e < 9; ++e) {
    float I = (e == 0 || e == 4 || e == 8) ? 1.0f : 0.0f;
    R[e] = I + s * K[e] + (1.0f - cc) * K2[e];
  }
}

static __device__ inline void mulGT(const float* G, const float* R,
                                    const float* rel, float* Go) {
#pragma unroll
  for (int m = 0; m < 3; ++m) {
#pragma unroll
    for (int n = 0; n < 3; ++n)
      Go[m * 4 + n] = G[m * 4 + 0] * R[0 * 3 + n] +
                      G[m * 4 + 1] * R[1 * 3 + n] +
                      G[m * 4 + 2] * R[2 * 3 + n];
    Go[m * 4 + 3] = G[m * 4 + 0] * rel[0] + G[m * 4 + 1] * rel[1] +
                    G[m * 4 + 2] * rel[2] + G[m * 4 + 3];
  }
}

static __device__ inline void storeA(const float* G, const float* jnt,
                                     float* out) {
#pragma unroll
  for (int m = 0; m < 3; ++m) {
    out[m * 4 + 0] = G[m * 4 + 0];
    out[m * 4 + 1] = G[m * 4 + 1];
    out[m * 4 + 2] = G[m * 4 + 2];
    out[m * 4 + 3] = G[m * 4 + 3] -
                     (G[m * 4 + 0] * jnt[0] + G[m * 4 + 1] * jnt[1] +
                      G[m * 4 + 2] * jnt[2]);
  }
}

__global__ void batch_kernel(const float* __restrict__ betasP,
                             const float* __restrict__ JSD,
                             const float* __restrict__ baseJ,
                             const float* __restrict__ pose,
                             const float* __restrict__ jaw,
                             float* __restrict__ poseFeat,
                             float* __restrict__ Amats) {
  int b = blockIdx.x * 64 + threadIdx.x;
  if (b >= B_) return;

  // joints = baseJ + JSD . betas
  float jnt[J_][3];
  const float* bb = betasP + b * KP;
  for (int j = 0; j < J_; ++j)
    for (int c = 0; c < 3; ++c) {
      const float* row = JSD + (j * 3 + c) * KP;
      float acc = baseJ[j * 3 + c];
      for (int l = 0; l < 150; ++l) acc += bb[l] * row[l];
      jnt[j][c] = acc;
    }

  // full pose: [global | neck=0 | jaw | eyes=0,0]
  float fp[J_][3];
  fp[0][0] = pose[b * 3 + 0]; fp[0][1] = pose[b * 3 + 1]; fp[0][2] = pose[b * 3 + 2];
  fp[1][0] = fp[1][1] = fp[1][2] = 0.0f;
  fp[2][0] = jaw[b * 3 + 0]; fp[2][1] = jaw[b * 3 + 1]; fp[2][2] = jaw[b * 3 + 2];
  fp[3][0] = fp[3][1] = fp[3][2] = 0.0f;
  fp[4][0] = fp[4][1] = fp[4][2] = 0.0f;

  float R[J_][9];
  for (int j = 0; j < J_; ++j) rodrigues1(fp[j][0], fp[j][1], fp[j][2], R[j]);

  // pose_feat = (R[1:] - I).flatten
  for (int j = 1; j < J_; ++j)
    for (int e = 0; e < 9; ++e) {
      float I = (e == 0 || e == 4 || e == 8) ? 1.0f : 0.0f;
      poseFeat[b * KPOSE + (j - 1) * 9 + e] = R[j][e] - I;
    }

  // relative joints (PARENTS = [-1,0,1,1,1])
  float rel[J_][3];
  for (int c = 0; c < 3; ++c) {
    rel[0][c] = jnt[0][c];
    rel[1][c] = jnt[1][c] - jnt[0][c];
    rel[2][c] = jnt[2][c] - jnt[1][c];
    rel[3][c] = jnt[3][c] - jnt[1][c];
    rel[4][c] = jnt[4][c] - jnt[1][c];
  }

  // chain (rows 0..2 of 4x4; row 3 = [0,0,0,1] implicit)
  float G0[12], G1[12], Gt[12];
  for (int m = 0; m < 3; ++m) {
    G0[m * 4 + 0] = R[0][m * 3 + 0];
    G0[m * 4 + 1] = R[0][m * 3 + 1];
    G0[m * 4 + 2] = R[0][m * 3 + 2];
    G0[m * 4 + 3] = rel[0][m];
  }
  float* Ab = Amats + b * 60;
  storeA(G0, jnt[0], Ab + 0);
  mulGT(G0, R[1], rel[1], G1);
  storeA(G1, jnt[1], Ab + 12);
  mulGT(G1, R[2], rel[2], Gt);
  storeA(Gt, jnt[2], Ab + 24);
  mulGT(G1, R[3], rel[3], Gt);
  storeA(Gt, jnt[3], Ab + 36);
  mulGT(G1, R[4], rel[4], Gt);
  storeA(Gt, jnt[4], Ab + 48);
}

// ---------------------------------------------------------------------------
// K3: fused WMMA GEMM (betas@SD^T + pose_feat@PD + vt) + LBS epilogue
//   block = 4 waves; wave tile = 16 batches x 48 cols (16 vertices x xyz)
//   Columns beyond NCOLS are clamped, not guarded: they belong exclusively to
//   the non-existent vertex 5023, which the epilogue never emits, so the
//   entire GEMM (both K segments + vt add) is branch-free / full-EXEC.
// ---------------------------------------------------------------------------
__global__ void __launch_bounds__(128)
gemm_lbs_kernel(const float* __restrict__ sd,       // (V*3, 150)
                const float* __restrict__ pd,       // (36, V*3)
                const float* __restrict__ vt,       // (V*3)
                const float* __restrict__ betasP,   // (B, 152)
                const float* __restrict__ poseFeat, // (B, 36)
                const float* __restrict__ Amats,    // (B, 5, 12)
                const float* __restrict__ lbsw,     // (V, 5)
                float* __restrict__ outV)           // (B, V, 3)
{
  __shared__ float sdT[NTILE * SD_STRIDE];           // 29.4 KB
  __shared__ float vpT[NWAVES * BT * NTILE];         // 12.3 KB

  const int tid   = threadIdx.x;
  const int wave  = tid >> 5;
  const int lane  = tid & 31;
  const int half  = lane >> 4;       // 0/1 lane half
  const int lq    = lane & 15;
  const int ksel  = 2 * half;        // K offset of this half (A/B frag layout)
  const int vtile = blockIdx.x;
  const int ncol0 = vtile * NTILE;   // global column base (v*3+c)
  const int b0    = (blockIdx.y * NWAVES + wave) * BT;

  // Hint the posedirs tile toward the caches while we stage SD into LDS.
  __builtin_prefetch(&pd[ncol0], 0, 1);
  __builtin_prefetch(&pd[18 * NCOLS + ncol0], 0, 1);

  // Stage shapedirs tile: sdT[n_local][l] = SD[(ncol0+n_local), l].
  // Data region goes through the async VMEM->LDS path (no VGPR roundtrip,
  // tracked by ASYNCcnt); K-pad and row-overhang entries are zeroed via DS.
  for (int idx = tid; idx < NTILE * KP; idx += 128) {
    int n = idx / KP, l = idx - n * KP;
    int ng = ncol0 + n;
    if (l < 150 && ng < NCOLS) {
#if USE_ASYNC_LDS
      __builtin_amdgcn_global_load_async_to_lds_b32(
          (g_as1_int*)(sd + ng * 150 + l),
          (l_as3_int*)(&sdT[n * SD_STRIDE + l]),
          0, 0);
#else
      sdT[n * SD_STRIDE + l] = sd[ng * 150 + l];
#endif
    } else {
      sdT[n * SD_STRIDE + l] = 0.0f;
    }
  }
#if USE_ASYNC_LDS
  __builtin_amdgcn_s_wait_asynccnt(0);
#endif
  __syncthreads();

  // Clamped global column per n-subtile (loop-invariant, dead-column safe).
  int ngc[3];
#pragma unroll
  for (int t = 0; t < 3; ++t) {
    int ng = ncol0 + t * 16 + lq;
    ngc[t] = (ng < NCOLS) ? ng : (NCOLS - 1);
  }

  v8f zero = {0.f, 0.f, 0.f, 0.f, 0.f, 0.f, 0.f, 0.f};
  v8f acc[3];
  acc[0] = zero; acc[1] = zero; acc[2] = zero;

  // --- K segment 1: betas (K=152 padded) x shapedirs^T (from LDS) ---
  const float* aRow = betasP + (b0 + lq) * KP;
  for (int k = 0; k < KP; k += 4) {
    v2f a;
    a.x = aRow[k + ksel];
    a.y = aRow[k + ksel + 1];
#pragma unroll
    for (int t = 0; t < 3; ++t) {
      const float* bp = &sdT[(t * 16 + lq) * SD_STRIDE + k + ksel];
      v2f bfr;
      bfr.x = bp[0];
      bfr.y = bp[1];
      acc[t] = wmma_f32x4(a, bfr, acc[t]);
    }
  }

  // --- K segment 2: pose_feat (K=36) x posedirs (coalesced, branch-free) ---
  const float* pRow = poseFeat + (b0 + lq) * KPOSE;
  for (int k = 0; k < KPOSE; k += 4) {
    v2f a;
    a.x = pRow[k + ksel];
    a.y = pRow[k + ksel + 1];
#pragma unroll
    for (int t = 0; t < 3; ++t) {
      const float* bp = pd + (k + ksel) * NCOLS + ngc[t];
      v2f bfr;
      bfr.x = bp[0];
      bfr.y = bp[NCOLS];
      acc[t] = wmma_f32x4(a, bfr, acc[t]);
    }
  }

  // Epilogue part 1: add v_template, spill v_posed tile to LDS
#pragma unroll
  for (int t = 0; t < 3; ++t) {
    float tv = vt[ngc[t]];
#pragma unroll
    for (int r = 0; r < 8; ++r) {
      int m = r + 8 * half;  // C/D layout: vgpr r, lane-half -> row
      vpT[(wave * BT + m) * NTILE + t * 16 + lq] = acc[t][r] + tv;
    }
  }
  __syncthreads();

  // Epilogue part 2: LBS. 256 (batch,vertex) pairs per wave, 8 per lane.
  for (int i = 0; i < 8; ++i) {
    int p  = i * 32 + lane;
    int bl = p >> 4;
    int vl = p & 15;
    int v  = vtile * 16 + vl;
    if (v >= V_) continue;
    int b = b0 + bl;

    const float* vp = &vpT[(wave * BT + bl) * NTILE + vl * 3];
    float x = vp[0], y = vp[1], z = vp[2];

    const float* Ab = Amats + b * 60;
    const float* w  = lbsw + v * J_;
    float T[12];
#pragma unroll
    for (int e = 0; e < 12; ++e) {
      float a2 = w[0] * Ab[e];
#pragma unroll
      for (int j = 1; j < J_; ++j) a2 += w[j] * Ab[j * 12 + e];
      T[e] = a2;
    }
    float ox = T[0] * x + T[1] * y + T[2]  * z + T[3];
    float oy = T[4] * x + T[5] * y + T[6]  * z + T[7];
    float oz = T[8] * x + T[9] * y + T[10] * z + T[11];

    float* o = outV + ((size_t)b * V_ + v) * 3;
    o[0] = ox; o[1] = oy; o[2] = oz;
  }
}

// ---------------------------------------------------------------------------
// K4: landmarks = sum_k bary[l,k] * vertices[b, faces[lmk_faces_idx[l],k], :]
// ---------------------------------------------------------------------------
__global__ void lmk_kernel(const float* __restrict__ outV,
                           const int* __restrict__ faces,
                           const int* __restrict__ lmkF,
                           const float* __restrict__ bary,
                           float* __restrict__ outL) {
  int b = blockIdx.x;
  int t = threadIdx.x;
  if (t >= L_ * 3) return;
  int l = t / 3, c = t - l * 3;
  int f = lmkF[l];
  float acc = 0.0f;
#pragma unroll
  for (int k = 0; k < 3; ++k) {
    int vtx = faces[f * 3 + k];
    acc += bary[l * 3 + k] * outV[((size_t)b * V_ + vtx) * 3 + c];
  }
  outL[((size_t)b * L_ + l) * 3 + c] = acc;
}

// ---------------------------------------------------------------------------
extern "C" void kernel_launch(void* const* d_in, const int* in_sizes, int n_in,
                              void* d_out, int out_size, void* d_ws,
                              size_t ws_size, hipStream_t stream) {
  const float* sp    = (const float*)d_in[0];   // shape_params (B,100)
  const float* ep    = (const float*)d_in[1];   // expr_params  (B,50)
  const float* pose  = (const float*)d_in[2];   // pose_params  (B,3)
  const float* jaw   = (const float*)d_in[3];   // jaw_params   (B,3)
  const float* vt    = (const float*)d_in[4];   // v_template   (V,3)
  const float* sd    = (const float*)d_in[5];   // shapedirs    (V,3,150)
  const float* pd    = (const float*)d_in[6];   // posedirs     (36, V*3)
  const float* JR    = (const float*)d_in[7];   // J_regressor  (5,V)
  const float* lbsw  = (const float*)d_in[8];   // lbs_weights  (V,5)
  const int*   faces = (const int*)d_in[9];     // faces        (F,3)
  const int*   lmkF  = (const int*)d_in[10];    // lmk_faces_idx(L,)
  const float* bary  = (const float*)d_in[11];  // lmk_bary     (L,3)

  float* outV = (float*)d_out;
  float* outL = outV + (size_t)B_ * V_ * 3;

  float* ws       = (float*)d_ws;
  float* betasP   = ws;                       // B*152
  float* JSD      = betasP + (size_t)B_ * KP; // 15*152
  float* baseJ    = JSD + 15 * KP;            // 16
  float* poseFeat = baseJ + 16;               // B*36
  float* Amats    = poseFeat + (size_t)B_ * KPOSE;  // B*60

  // K1: padded betas
  betas_pad_kernel<<<(B_ * KP + 255) / 256, 256, 0, stream>>>(sp, ep, betasP);
  // K0: joint regressor folded into shapedirs
  jsd_kernel<<<J_, 512, 0, stream>>>(JR, sd, vt, JSD, baseJ);
  // K2: per-batch joints / rotations / chain / A matrices / pose features
  batch_kernel<<<B_ / 64, 64, 0, stream>>>(betasP, JSD, baseJ, pose, jaw,
                                           poseFeat, Amats);
  // K3: fused WMMA blendshape+pose GEMM and LBS
  dim3 grid3((V_ + 15) / 16, B_ / (NWAVES * BT));
  gemm_lbs_kernel<<<grid3, NWAVES * 32, 0, stream>>>(sd, pd, vt, betasP,
                                                     poseFeat, Amats, lbsw,
                                                     outV);
  // K4: landmarks
  lmk_kernel<<<B_, 224, 0, stream>>>(outV, faces, lmkF, bary, outL);
}